// ouroboros_68350109548825
// MI455X (gfx1250) — compile-verified
//
#include <hip/hip_runtime.h>
#include <hip/hip_bf16.h>

// ---------------- Types for WMMA fragments (CDNA5 fp32 WMMA 16x16x4) -------
typedef __attribute__((ext_vector_type(2))) float v2f;
typedef __attribute__((ext_vector_type(8))) float v8f;

static __device__ __forceinline__ v8f wmma4(v2f a, v2f b, v8f c) {
  // V_WMMA_F32_16X16X4_F32 : D = A(16x4) * B(4x16) + C(16x16), fp32 throughout
  return __builtin_amdgcn_wmma_f32_16x16x4_f32(
      /*neg_a=*/false, a, /*neg_b=*/false, b,
      /*c_mod=*/(short)0, c, /*reuse_a=*/false, /*reuse_b=*/false);
}

static __device__ __forceinline__ float sigmoidf_(float v) {
  return 1.0f / (1.0f + __expf(-v));
}

// ---------------- CDNA5 async global->LDS path ------------------------------
#if __has_builtin(__builtin_amdgcn_global_load_async_to_lds_b128)
#define HAVE_ASYNC_LDS 1
typedef int v4i __attribute__((ext_vector_type(4)));
typedef __attribute__((address_space(1))) v4i gv4i;   // global int4 (AS1)
typedef __attribute__((address_space(3))) v4i lv4i;   // LDS int4 (AS3)
#else
#define HAVE_ASYNC_LDS 0
#endif

static __device__ __forceinline__ void wait_async0() {
#if HAVE_ASYNC_LDS
#if __has_builtin(__builtin_amdgcn_s_wait_asynccnt)
  __builtin_amdgcn_s_wait_asynccnt(0);
#else
  asm volatile("s_wait_asynccnt 0" ::: "memory");
#endif
#endif
}

// ---------------- Problem constants ----------------------------------------
#define BSZ   2
#define LSEQ  1024
#define DM    1024
#define ED    1024
#define NST   16
#define DTR   64
#define DCONV 4
#define DCTRL 512
#define DDATA 512
#define MROWS (BSZ * LSEQ)      // 2048

// ---------------- Generic fp32 WMMA GEMM:  C = epi(A @ W^T + bias + resid) --
// A: (M x K) row-major, row stride lda
// W: (N x K) row-major, row stride ldw    (so B[k][n] = W[n][k])
// C: (M x N) row-major, row stride ldc
// epi: 0 = none, 1 = silu, 2 = softplus
#define BM 128
#define BN 64
#define BK 16
#define BKP 20   // row pitch: 80B, 16B-aligned rows for async b128, conflict-free

__global__ __launch_bounds__(256) void gemm_f32_wmma(
    const float* __restrict__ A, int lda,
    const float* __restrict__ W, int ldw,
    float* __restrict__ C, int ldc,
    int M, int N, int K,
    const float* __restrict__ bias,
    const float* __restrict__ resid,
    int epi)
{
  // double-buffered tiles; both stored row-pitch BKP, rows hold 16 K-floats
  __shared__ float As[2][BM][BKP];   // [m][k]
  __shared__ float Bs[2][BN][BKP];   // [n][k]  (== row-major slice of W)

  const int m0 = blockIdx.y * BM;
  const int n0 = blockIdx.x * BN;
  const int t    = threadIdx.x;
  const int wave = t >> 5;
  const int lane = t & 31;
  const int lm = lane & 15;          // M (for A) / N (for B) index within tile
  const int lh = lane >> 4;          // selects K pair / M half
  const int wm = (wave & 3) * 32;    // 4 waves along M
  const int wn = (wave >> 2) * 32;   // 2 waves along N

  // staging coordinates (each thread moves 2 A-chunks + 1 B-chunk of 16B)
  const int am0 = (t + 0)   >> 2, akq0 = ((t + 0)   & 3) << 2;
  const int am1 = (t + 256) >> 2, akq1 = ((t + 256) & 3) << 2;
  const int bn  = t >> 2,         bkq  = (t & 3) << 2;

  auto stage = [&](int buf, int k0) {
#if HAVE_ASYNC_LDS
    __builtin_amdgcn_global_load_async_to_lds_b128(
        (gv4i*)(A + (size_t)(m0 + am0) * lda + k0 + akq0),
        (lv4i*)&As[buf][am0][akq0], 0, 0);
    __builtin_amdgcn_global_load_async_to_lds_b128(
        (gv4i*)(A + (size_t)(m0 + am1) * lda + k0 + akq1),
        (lv4i*)&As[buf][am1][akq1], 0, 0);
    if (n0 + bn < N) {
      __builtin_amdgcn_global_load_async_to_lds_b128(
          (gv4i*)(W + (size_t)(n0 + bn) * ldw + k0 + bkq),
          (lv4i*)&Bs[buf][bn][bkq], 0, 0);
    }
    // out-of-range B columns stay stale in LDS; their products are never stored
#else
    {
      float4 v = *(const float4*)(A + (size_t)(m0 + am0) * lda + k0 + akq0);
      As[buf][am0][akq0 + 0] = v.x; As[buf][am0][akq0 + 1] = v.y;
      As[buf][am0][akq0 + 2] = v.z; As[buf][am0][akq0 + 3] = v.w;
      v = *(const float4*)(A + (size_t)(m0 + am1) * lda + k0 + akq1);
      As[buf][am1][akq1 + 0] = v.x; As[buf][am1][akq1 + 1] = v.y;
      As[buf][am1][akq1 + 2] = v.z; As[buf][am1][akq1 + 3] = v.w;
    }
    {
      float4 v = make_float4(0.f, 0.f, 0.f, 0.f);
      if (n0 + bn < N) v = *(const float4*)(W + (size_t)(n0 + bn) * ldw + k0 + bkq);
      Bs[buf][bn][bkq + 0] = v.x; Bs[buf][bn][bkq + 1] = v.y;
      Bs[buf][bn][bkq + 2] = v.z; Bs[buf][bn][bkq + 3] = v.w;
    }
#endif
  };

  v8f acc[2][2];
  #pragma unroll
  for (int i = 0; i < 2; ++i)
    #pragma unroll
    for (int j = 0; j < 2; ++j)
      acc[i][j] = (v8f)(0.0f);

  const int nk = K / BK;
  stage(0, 0);                     // prologue: fill buffer 0

  for (int kc = 0; kc < nk; ++kc) {
    const int cur = kc & 1;
    wait_async0();                 // my async copies (into `cur`) landed
    __syncthreads();               // everyone's copies landed; prior reads done
    if (kc + 1 < nk) stage(cur ^ 1, (kc + 1) * BK);   // prefetch overlaps WMMA

    #pragma unroll
    for (int s = 0; s < 4; ++s) {
      const int ka = 4 * s + 2 * lh;     // K pair this lane-half supplies
      v2f a0, a1, b0, b1;
      a0.x = As[cur][wm + lm][ka];        a0.y = As[cur][wm + lm][ka + 1];
      a1.x = As[cur][wm + 16 + lm][ka];   a1.y = As[cur][wm + 16 + lm][ka + 1];
      b0.x = Bs[cur][wn + lm][ka];        b0.y = Bs[cur][wn + lm][ka + 1];
      b1.x = Bs[cur][wn + 16 + lm][ka];   b1.y = Bs[cur][wn + 16 + lm][ka + 1];
      acc[0][0] = wmma4(a0, b0, acc[0][0]);
      acc[0][1] = wmma4(a0, b1, acc[0][1]);
      acc[1][0] = wmma4(a1, b0, acc[1][0]);
      acc[1][1] = wmma4(a1, b1, acc[1][1]);
    }
  }

  // --- epilogue: C layout per ISA: acc[r] -> M = r + 8*lh, N = lm ---------
  #pragma unroll
  for (int i = 0; i < 2; ++i) {
    #pragma unroll
    for (int j = 0; j < 2; ++j) {
      const int n = n0 + wn + j * 16 + lm;
      if (n < N) {
        #pragma unroll
        for (int r = 0; r < 8; ++r) {
          const int m = m0 + wm + i * 16 + r + 8 * lh;
          float v = acc[i][j][r];
          if (bias)  v += bias[n];
          if (resid) v += resid[(size_t)m * ldc + n];
          if (epi == 1)      v = v * sigmoidf_(v);                       // silu
          else if (epi == 2) v = (v > 20.f) ? v : log1pf(__expf(v));     // softplus
          C[(size_t)m * ldc + n] = v;
        }
      }
    }
  }
}

// ---------------- RMSNorm: one block per row (D = 1024) ---------------------
__global__ __launch_bounds__(256) void rmsnorm_kernel(
    const float* __restrict__ u, const float* __restrict__ w,
    float* __restrict__ out)
{
  __shared__ float red[256];
  const int row = blockIdx.x;
  const int t = threadIdx.x;
  const float* ur = u + (size_t)row * DM;
  float s = 0.f;
  #pragma unroll
  for (int c = t; c < DM; c += 256) { float v = ur[c]; s += v * v; }
  red[t] = s;
  __syncthreads();
  for (int off = 128; off > 0; off >>= 1) {
    if (t < off) red[t] += red[t + off];
    __syncthreads();
  }
  const float rms = rsqrtf(red[0] * (1.0f / DM) + 1e-5f);
  #pragma unroll
  for (int c = t; c < DM; c += 256)
    out[(size_t)row * DM + c] = ur[c] * rms * w[c];
}

// ---------------- Causal depthwise conv (k=4) + bias + silu ------------------
__global__ __launch_bounds__(256) void conv_silu_kernel(
    const float* __restrict__ xz,        // (M, 2*ED); first ED cols are xc_raw
    const float* __restrict__ conv_w,    // (ED, 4)
    const float* __restrict__ conv_b,    // (ED,)
    float* __restrict__ xc)              // (M, ED)
{
  const int idx = blockIdx.x * 256 + threadIdx.x;   // over B*L*ED
  if (idx >= MROWS * ED) return;
  const int d   = idx & (ED - 1);
  const int row = idx >> 10;
  const int l   = row & (LSEQ - 1);
  const int b   = row >> 10;
  float acc = conv_b[d];
  #pragma unroll
  for (int j = 0; j < DCONV; ++j) {
    const int ll = l - (DCONV - 1) + j;
    if (ll >= 0)
      acc += conv_w[d * DCONV + j] * xz[((size_t)(b * LSEQ + ll)) * (2 * ED) + d];
  }
  xc[idx] = acc * sigmoidf_(acc);
}

// ---------------- Selective scan fused with D*xc + silu(z) gating -----------
__global__ __launch_bounds__(256) void scan_kernel(
    const float* __restrict__ delta,   // (M, ED)
    const float* __restrict__ xc,      // (M, ED)
    const float* __restrict__ dbc,     // (M, 96): [0:64]=dt_in, [64:80]=B, [80:96]=C
    const float* __restrict__ xz,      // (M, 2*ED): z in second half
    const float* __restrict__ A_log,   // (ED, N)
    const float* __restrict__ Dv,      // (ED,)
    float* __restrict__ yv)            // (M, ED)
{
  const int gid = blockIdx.x * 256 + threadIdx.x;   // B*ED = 2048 threads
  if (gid >= BSZ * ED) return;
  const int b = gid >> 10;
  const int d = gid & (ED - 1);

  float A[NST], h[NST];
  #pragma unroll
  for (int n = 0; n < NST; ++n) { A[n] = -__expf(A_log[d * NST + n]); h[n] = 0.f; }
  const float Dvd = Dv[d];

  for (int l = 0; l < LSEQ; ++l) {
    const size_t row = (size_t)b * LSEQ + l;
    const float dl = delta[row * ED + d];
    const float xv = xc[row * ED + d];
    const float dx = dl * xv;
    const float* bc = dbc + row * (DTR + 2 * NST);
    float acc = 0.f;
    #pragma unroll
    for (int n = 0; n < NST; ++n) {
      const float hn = __expf(dl * A[n]) * h[n] + dx * bc[DTR + n];
      h[n] = hn;
      acc += hn * bc[DTR + NST + n];
    }
    const float z = xz[row * (2 * ED) + ED + d];
    yv[row * ED + d] = (acc + Dvd * xv) * (z * sigmoidf_(z));
  }
}

// ---------------- Concat builders -------------------------------------------
__global__ __launch_bounds__(256) void build_u1_kernel(
    const float* __restrict__ x, const float* __restrict__ y, float* __restrict__ u)
{
  const int idx = blockIdx.x * 256 + threadIdx.x;   // over B*L*512
  if (idx >= MROWS * DDATA) return;
  const int row = idx >> 9, c = idx & (DDATA - 1);
  const float xv = x[idx], yv_ = y[idx];
  u[(size_t)row * DM + c]         = yv_ - xv;   // dy
  u[(size_t)row * DM + DDATA + c] = yv_;        // y
}

__global__ __launch_bounds__(256) void build_u2_kernel(
    const float* __restrict__ sp, const float* __restrict__ x,
    const int* __restrict__ mask_ind, float* __restrict__ u)
{
  const int idx = blockIdx.x * 256 + threadIdx.x;   // over B*L*512
  if (idx >= MROWS * DDATA) return;
  const int row = idx >> 9, c = idx & (DDATA - 1);
  const int l = row & (LSEQ - 1);
  u[(size_t)row * DM + c] = sp[idx];
  float xv = x[idx];
  if (l == *mask_ind) xv = 0.f;
  u[(size_t)row * DM + DDATA + c] = xv;
}

// ---------------- Host-side orchestration -----------------------------------
static void launch_gemm(const float* A, int lda, const float* W, int ldw,
                        float* C, int ldc, int M, int N, int K,
                        const float* bias, const float* resid, int epi,
                        hipStream_t s)
{
  dim3 grid((N + BN - 1) / BN, M / BM);
  gemm_f32_wmma<<<grid, 256, 0, s>>>(A, lda, W, ldw, C, ldc, M, N, K, bias, resid, epi);
}

static void run_layer(const float* uin, float* cout,
                      const float* norm, const float* in_w,
                      const float* conv_w, const float* conv_b,
                      const float* xp_w, const float* dt_w, const float* dt_b,
                      const float* A_log, const float* Dv, const float* out_w,
                      float* XN, float* XZ, float* XC, float* DBC, float* DELTA,
                      float* YV, hipStream_t s)
{
  rmsnorm_kernel<<<MROWS, 256, 0, s>>>(uin, norm, XN);
  launch_gemm(XN, DM, in_w, DM, XZ, 2 * ED, MROWS, 2 * ED, DM, nullptr, nullptr, 0, s);
  conv_silu_kernel<<<(MROWS * ED) / 256, 256, 0, s>>>(XZ, conv_w, conv_b, XC);
  launch_gemm(XC, ED, xp_w, ED, DBC, DTR + 2 * NST, MROWS, DTR + 2 * NST, ED,
              nullptr, nullptr, 0, s);
  launch_gemm(DBC, DTR + 2 * NST, dt_w, DTR, DELTA, ED, MROWS, ED, DTR,
              dt_b, nullptr, /*softplus*/2, s);
  scan_kernel<<<(BSZ * ED) / 256, 256, 0, s>>>(DELTA, XC, DBC, XZ, A_log, Dv, YV);
  launch_gemm(YV, ED, out_w, ED, cout, DM, MROWS, DM, ED, nullptr, uin, 0, s);
}

extern "C" void kernel_launch(void* const* d_in, const int* in_sizes, int n_in,
                              void* d_out, int out_size, void* d_ws, size_t ws_size,
                              hipStream_t stream)
{
  const float* x = (const float*)d_in[0];
  const float* y = (const float*)d_in[1];
  const float* cn[10]; for (int i = 0; i < 10; ++i) cn[i] = (const float*)d_in[2 + i];
  const float* dn[10]; for (int i = 0; i < 10; ++i) dn[i] = (const float*)d_in[12 + i];
  const float* cp_w   = (const float*)d_in[22];
  const float* cp_b   = (const float*)d_in[23];
  const int*   maskp  = (const int*)d_in[24];

  float* out  = (float*)d_out;
  float* yhat = out;                               // (2,1024,1024)
  float* sp   = out + (size_t)MROWS * DM;          // (2,1024,512)

  float* ws    = (float*)d_ws;
  float* U0    = ws;                   // 2,097,152
  float* U1    = ws + 2097152;         // 2,097,152
  float* XN    = ws + 4194304;         // 2,097,152  (aliased as YV: XN dead after in-proj)
  float* XZ    = ws + 6291456;         // 4,194,304
  float* XC    = ws + 10485760;        // 2,097,152
  float* DBC   = ws + 12582912;        //   196,608
  float* DELTA = ws + 12779520;        // 2,097,152

  const int st[10] = {DM, 2 * ED * DM, ED * DCONV, ED, (DTR + 2 * NST) * ED,
                      ED * DTR, ED, ED * NST, ED, DM * ED};

  const int elems512 = MROWS * DDATA;
  build_u1_kernel<<<elems512 / 256, 256, 0, stream>>>(x, y, U0);

  // --- control-path mamba (2 layers): U0 -> U1 -> U0 -----------------------
  run_layer(U0, U1,
            cn[0], cn[1], cn[2], cn[3], cn[4], cn[5], cn[6], cn[7], cn[8], cn[9],
            XN, XZ, XC, DBC, DELTA, XN, stream);
  run_layer(U1, U0,
            cn[0]+st[0], cn[1]+st[1], cn[2]+st[2], cn[3]+st[3], cn[4]+st[4],
            cn[5]+st[5], cn[6]+st[6], cn[7]+st[7], cn[8]+st[8], cn[9]+st[9],
            XN, XZ, XC, DBC, DELTA, XN, stream);

  // sp = silu(sp_pre @ cp_w^T + cp_b), written straight into d_out
  launch_gemm(U0, DM, cp_w, DM, sp, DCTRL, MROWS, DCTRL, DM, cp_b, nullptr,
              /*silu*/1, stream);

  // u2 = concat(sp, x with row mask_ind zeroed)
  build_u2_kernel<<<elems512 / 256, 256, 0, stream>>>(sp, x, maskp, U1);

  // --- data-path mamba (2 layers): U1 -> U0 -> yhat (d_out) ----------------
  run_layer(U1, U0,
            dn[0], dn[1], dn[2], dn[3], dn[4], dn[5], dn[6], dn[7], dn[8], dn[9],
            XN, XZ, XC, DBC, DELTA, XN, stream);
  run_layer(U0, yhat,
            dn[0]+st[0], dn[1]+st[1], dn[2]+st[2], dn[3]+st[3], dn[4]+st[4],
            dn[5]+st[5], dn[6]+st[6], dn[7]+st[7], dn[8]+st[8], dn[9]+st[9],
            XN, XZ, XC, DBC, DELTA, XN, stream);
}